// SelfAttentionWithPagedCache_26817775796986
// MI455X (gfx1250) — compile-verified
//
#include <hip/hip_runtime.h>
#include <cstdint>

#define D_MODEL   1024
#define N_HEADS   16
#define HEAD_DIM  64
#define S_NEW     16
#define PAST_LEN  32768
#define CHUNK_KEYS 512
#define NCH_PAST  (PAST_LEN / CHUNK_KEYS)   // 64
#define NCH       (NCH_PAST + 1)            // 65 (last chunk = 16 new keys, causal)
#define TILES_PER_CHUNK (CHUNK_KEYS / 16)   // 32
#define LDS_K_STRIDE 68                     // 64 + 4 pad: conflict-free b64 reads
#define LDS_P_STRIDE 18                     // 16 + 2 pad: 8B-aligned, conflict-free

typedef float v2f __attribute__((ext_vector_type(2)));
typedef float v8f __attribute__((ext_vector_type(8)));

__device__ __forceinline__ v8f wmma_f32_16x16x4(v2f a, v2f b, v8f c) {
  // (neg_a, A, neg_b, B, c_mod, C, reuse_a, reuse_b)
  return __builtin_amdgcn_wmma_f32_16x16x4_f32(false, a, false, b, (short)0, c,
                                               false, false);
}

__device__ __forceinline__ uint32_t lds_off(const void* p) {
  // generic pointer to __shared__: low 32 bits are the wave-relative LDS offset
  return (uint32_t)(uintptr_t)p;
}

// Issue async copy of two 16x64 f32 tiles (row stride D_MODEL) into padded LDS
// tiles (row stride LDS_K_STRIDE). 16 x global_load_async_to_lds_b128, ASYNCcnt +=16.
__device__ __forceinline__ void async_stage_16x64x2(const float* __restrict__ pa,
                                                    const float* __restrict__ pb,
                                                    uint32_t la, uint32_t lb,
                                                    int lane) {
  const int j0   = lane >> 4;   // 0..1
  const int part = lane & 15;   // 0..15 (float4 chunk within row)
  #pragma unroll
  for (int it = 0; it < 8; ++it) {
    const int j = it * 2 + j0;
    const uint32_t ofl = (uint32_t)((j * LDS_K_STRIDE + part * 4) * 4);
    const uint64_t ga  = (uint64_t)(uintptr_t)(pa + (size_t)j * D_MODEL + part * 4);
    const uint64_t gb  = (uint64_t)(uintptr_t)(pb + (size_t)j * D_MODEL + part * 4);
    asm volatile("global_load_async_to_lds_b128 %0, %1, off"
                 :: "v"(la + ofl), "v"(ga) : "memory");
    asm volatile("global_load_async_to_lds_b128 %0, %1, off"
                 :: "v"(lb + ofl), "v"(gb) : "memory");
  }
}

__device__ __forceinline__ void wait_async_le16() {
  asm volatile("s_wait_asynccnt 0x10" ::: "memory");
}
__device__ __forceinline__ void wait_async_0() {
  asm volatile("s_wait_asynccnt 0x0" ::: "memory");
}

// ---------------------------------------------------------------------------
// y[s, n0..n0+15] = sum_k x[s,k] * W[n0+n, k]   (y = x @ W^T), 16 rows of x.
// One wave per 16x16 tile; K staged via double-buffered async-to-LDS.
// ---------------------------------------------------------------------------
__device__ __forceinline__ void gemm_tile_16(const float* __restrict__ x,
                                             const float* __restrict__ W,
                                             float* __restrict__ y,
                                             int n0, float* xs, float* wsh) {
  const int lane = threadIdx.x & 31;
  const int sn   = lane & 15;   // A: m (row of x) / B: n (col of y)
  const int hi   = lane >> 4;
  const int koff = hi * 2;      // f32 A/B layout: lanes 0-15 hold K={0,1}, 16-31 K={2,3}
  const int BUF  = 16 * LDS_K_STRIDE;

  const float* Wr = W + (size_t)n0 * D_MODEL;
  const uint32_t xso = lds_off(xs), wso = lds_off(wsh);

  v8f acc = {0.f, 0.f, 0.f, 0.f, 0.f, 0.f, 0.f, 0.f};

  async_stage_16x64x2(x, Wr, xso, wso, lane);   // chunk 0 -> buffer 0
  for (int c = 0; c < 16; ++c) {                // 16 chunks of K=64
    const int cur = c & 1;
    if (c + 1 < 16) {
      const int nk0 = (c + 1) * 64;
      async_stage_16x64x2(x + nk0, Wr + nk0,
                          xso + (uint32_t)(((c + 1) & 1) * BUF * 4),
                          wso + (uint32_t)(((c + 1) & 1) * BUF * 4), lane);
      wait_async_le16();                        // chunk c landed
    } else {
      wait_async_0();
    }
    const float* xc = xs  + cur * BUF;
    const float* wc = wsh + cur * BUF;
    #pragma unroll
    for (int kk = 0; kk < 64; kk += 4) {
      v2f a = *(const v2f*)(xc + sn * LDS_K_STRIDE + kk + koff);
      v2f b = *(const v2f*)(wc + sn * LDS_K_STRIDE + kk + koff);
      acc = wmma_f32_16x16x4(a, b, acc);
    }
  }
  #pragma unroll
  for (int r = 0; r < 8; ++r)
    y[(r + 8 * hi) * D_MODEL + n0 + sn] = acc[r];  // C layout: m=r+8*hi, n=lane%16
}

__global__ void __launch_bounds__(32)
qkv_kernel(const float* __restrict__ x,
           const float* __restrict__ Wq, const float* __restrict__ Wk,
           const float* __restrict__ Wv,
           float* __restrict__ q, float* __restrict__ k, float* __restrict__ v) {
  __shared__ alignas(16) float xs[2 * 16 * LDS_K_STRIDE];
  __shared__ alignas(16) float wsh[2 * 16 * LDS_K_STRIDE];
  const int bx  = blockIdx.x;          // 192 blocks: 3 matrices x 64 tiles
  const int mat = bx >> 6;
  const int n0  = (bx & 63) << 4;
  const float* W = (mat == 0) ? Wq : (mat == 1) ? Wk : Wv;
  float*       y = (mat == 0) ? q  : (mat == 1) ? k  : v;
  gemm_tile_16(x, W, y, n0, xs, wsh);
}

__global__ void __launch_bounds__(32)
oproj_kernel(const float* __restrict__ ao, const float* __restrict__ Wo,
             float* __restrict__ out) {
  __shared__ alignas(16) float xs[2 * 16 * LDS_K_STRIDE];
  __shared__ alignas(16) float wsh[2 * 16 * LDS_K_STRIDE];
  gemm_tile_16(ao, Wo, out, (int)blockIdx.x << 4, xs, wsh);
}

// ---------------------------------------------------------------------------
// Flash-attention partial over one (head, key-chunk). One wave per block.
// Sᵀ = K_tile(16x64) @ Qᵀ(64x16): softmax state is per-lane (lane = query).
// Oᵀ(64x16) = Vᵀ @ Pᵀ accumulated as 4 WMMA tiles. K/V stream in via
// double-buffered global_load_async_to_lds_b128. Partials (Oᵀ, m, l) -> ws.
// ---------------------------------------------------------------------------
__global__ void __launch_bounds__(32)
attn_partial_kernel(const float* __restrict__ q,
                    const float* __restrict__ past_k, const float* __restrict__ past_v,
                    const float* __restrict__ k_new,  const float* __restrict__ v_new,
                    float* __restrict__ o_part, float* __restrict__ m_part,
                    float* __restrict__ l_part) {
  __shared__ alignas(16) float kt[2 * 16 * LDS_K_STRIDE];
  __shared__ alignas(16) float vt[2 * 16 * LDS_K_STRIDE];
  __shared__ float pt[16 * LDS_P_STRIDE];
  const int BUF = 16 * LDS_K_STRIDE;

  const int chunk = blockIdx.x;          // 0..NCH-1
  const int h     = blockIdx.y;          // head
  const int lane  = threadIdx.x & 31;
  const int sn    = lane & 15;           // query index s (dup across halves)
  const int hi    = lane >> 4;
  const int koff  = hi * 2;

  const bool is_new = (chunk == NCH_PAST);
  const float* Kp = is_new ? (k_new + h * HEAD_DIM)
                           : (past_k + (size_t)chunk * CHUNK_KEYS * D_MODEL + h * HEAD_DIM);
  const float* Vp = is_new ? (v_new + h * HEAD_DIM)
                           : (past_v + (size_t)chunk * CHUNK_KEYS * D_MODEL + h * HEAD_DIM);
  const int ntiles = is_new ? 1 : TILES_PER_CHUNK;
  const uint32_t kto = lds_off(kt), vto = lds_off(vt);

  // Preload Q for this head in WMMA B layout: B[k=d, n=s].
  v2f qb[16];
  #pragma unroll
  for (int d0 = 0; d0 < HEAD_DIM; d0 += 4)
    qb[d0 >> 2] = *(const v2f*)(q + sn * D_MODEL + h * HEAD_DIM + d0 + koff);

  float m_run = -1e30f, l_run = 0.f;
  v8f acc[4];
  #pragma unroll
  for (int t = 0; t < 4; ++t) acc[t] = (v8f){0.f,0.f,0.f,0.f,0.f,0.f,0.f,0.f};

  async_stage_16x64x2(Kp, Vp, kto, vto, lane);   // tile 0 -> buffer 0
  for (int i = 0; i < ntiles; ++i) {
    const int cur = i & 1;
    if (i + 1 < ntiles) {
      async_stage_16x64x2(Kp + (size_t)(i + 1) * 16 * D_MODEL,
                          Vp + (size_t)(i + 1) * 16 * D_MODEL,
                          kto + (uint32_t)(((i + 1) & 1) * BUF * 4),
                          vto + (uint32_t)(((i + 1) & 1) * BUF * 4), lane);
      wait_async_le16();                         // tile i landed
    } else {
      wait_async_0();
    }
    const float* ktc = kt + cur * BUF;
    const float* vtc = vt + cur * BUF;

    // ---- Sᵀ = K_tile @ Qᵀ  (m=key j, n=query s), K-dim = 64
    v8f st = {0.f,0.f,0.f,0.f,0.f,0.f,0.f,0.f};
    #pragma unroll
    for (int d0 = 0; d0 < HEAD_DIM; d0 += 4) {
      v2f a = *(const v2f*)(ktc + sn * LDS_K_STRIDE + d0 + koff);  // A[m=j][k=d]
      st = wmma_f32_16x16x4(a, qb[d0 >> 2], st);
    }

    // ---- scale + causal mask (new-keys chunk only: key j > query s masked)
    float sc[8];
    #pragma unroll
    for (int r = 0; r < 8; ++r) sc[r] = st[r] * 0.125f;  // 1/sqrt(64)
    if (is_new) {
      #pragma unroll
      for (int r = 0; r < 8; ++r)
        if ((r + 8 * hi) > sn) sc[r] = -1e30f;
    }

    // ---- online softmax: per-lane (per-query) state, one half-swap shuffle
    float tmax = sc[0];
    #pragma unroll
    for (int r = 1; r < 8; ++r) tmax = fmaxf(tmax, sc[r]);
    tmax = fmaxf(tmax, __shfl_xor(tmax, 16, 32));
    float m_new = fmaxf(m_run, tmax);
    float alpha = __expf(m_run - m_new);
    float p[8], psum = 0.f;
    #pragma unroll
    for (int r = 0; r < 8; ++r) { p[r] = __expf(sc[r] - m_new); psum += p[r]; }
    psum += __shfl_xor(psum, 16, 32);
    l_run = l_run * alpha + psum;
    m_run = m_new;
    #pragma unroll
    for (int t = 0; t < 4; ++t)
      #pragma unroll
      for (int r = 0; r < 8; ++r) acc[t][r] *= alpha;

    // ---- Pᵀ: C layout -> LDS as P[s][j], re-read in B layout
    #pragma unroll
    for (int r = 0; r < 8; ++r) pt[sn * LDS_P_STRIDE + r + 8 * hi] = p[r];
    __syncthreads();
    v2f pb[4];
    #pragma unroll
    for (int kk = 0; kk < 16; kk += 4)
      pb[kk >> 2] = *(const v2f*)(pt + sn * LDS_P_STRIDE + kk + koff);  // B[k=j][n=s]

    // ---- Oᵀ += Vᵀ @ Pᵀ : 4 d-tiles of 16, K-dim = 16 keys
    #pragma unroll
    for (int t = 0; t < 4; ++t) {
      #pragma unroll
      for (int kk = 0; kk < 16; kk += 4) {
        v2f a;                                 // A[m = d-16t][k = j]
        a.x = vtc[(kk + koff)     * LDS_K_STRIDE + t * 16 + sn];
        a.y = vtc[(kk + koff + 1) * LDS_K_STRIDE + t * 16 + sn];
        acc[t] = wmma_f32_16x16x4(a, pb[kk >> 2], acc[t]);
      }
    }
    __syncthreads();
  }

  // ---- write partials: Oᵀ[64][16], m[16], l[16]
  const int pidx = h * NCH + chunk;
  #pragma unroll
  for (int t = 0; t < 4; ++t)
    #pragma unroll
    for (int r = 0; r < 8; ++r)
      o_part[((size_t)pidx * HEAD_DIM + t * 16 + r + 8 * hi) * 16 + sn] = acc[t][r];
  if (lane < 16) {
    m_part[pidx * 16 + lane] = m_run;
    l_part[pidx * 16 + lane] = l_run;
  }
}

// ---------------------------------------------------------------------------
// Combine split-T partials: ao[s, h*64+d] = sum_c e^{m_c-M} Oᵀ_c[d][s] / sum_c e^{m_c-M} l_c
// ---------------------------------------------------------------------------
__global__ void attn_reduce_kernel(const float* __restrict__ o_part,
                                   const float* __restrict__ m_part,
                                   const float* __restrict__ l_part,
                                   float* __restrict__ ao) {
  const int h = blockIdx.x;
  for (int idx = threadIdx.x; idx < HEAD_DIM * 16; idx += blockDim.x) {
    const int d = idx >> 4, s = idx & 15;
    float M = -1e30f;
    for (int c = 0; c < NCH; ++c)
      M = fmaxf(M, m_part[(h * NCH + c) * 16 + s]);
    float num = 0.f, den = 0.f;
    for (int c = 0; c < NCH; ++c) {
      const int pidx = h * NCH + c;
      float a = __expf(m_part[pidx * 16 + s] - M);
      num += a * o_part[((size_t)pidx * HEAD_DIM + d) * 16 + s];
      den += a * l_part[pidx * 16 + s];
    }
    ao[s * D_MODEL + h * HEAD_DIM + d] = num / den;
  }
}

// ---------------------------------------------------------------------------
extern "C" void kernel_launch(void* const* d_in, const int* in_sizes, int n_in,
                              void* d_out, int out_size, void* d_ws, size_t ws_size,
                              hipStream_t stream) {
  (void)in_sizes; (void)n_in; (void)out_size; (void)ws_size;
  const float* x      = (const float*)d_in[0];
  const float* past_k = (const float*)d_in[1];
  const float* past_v = (const float*)d_in[2];
  const float* Wq     = (const float*)d_in[3];
  const float* Wk     = (const float*)d_in[4];
  const float* Wv     = (const float*)d_in[5];
  const float* Wo     = (const float*)d_in[6];
  float* out = (float*)d_out;

  float* ws     = (float*)d_ws;
  float* q_ws   = ws;                 // 16x1024
  float* k_ws   = q_ws + S_NEW * D_MODEL;
  float* v_ws   = k_ws + S_NEW * D_MODEL;
  float* ao_ws  = v_ws + S_NEW * D_MODEL;
  float* o_part = ao_ws + S_NEW * D_MODEL;                    // 16*65*64*16 floats
  float* m_part = o_part + (size_t)N_HEADS * NCH * HEAD_DIM * 16;
  float* l_part = m_part + N_HEADS * NCH * 16;

  qkv_kernel<<<192, 32, 0, stream>>>(x, Wq, Wk, Wv, q_ws, k_ws, v_ws);
  attn_partial_kernel<<<dim3(NCH, N_HEADS), 32, 0, stream>>>(
      q_ws, past_k, past_v, k_ws, v_ws, o_part, m_part, l_part);
  attn_reduce_kernel<<<N_HEADS, 256, 0, stream>>>(o_part, m_part, l_part, ao_ws);
  oproj_kernel<<<64, 32, 0, stream>>>(ao_ws, Wo, out);
}